// HungarianMatcher_43619687858432
// MI455X (gfx1250) — compile-verified
//
#include <hip/hip_runtime.h>

// CDNA5 / gfx1250: wave32, WMMA. Exact gather-as-matmul via f32 WMMA:
//   cost_class[q,m] = -(probs(16xK) x onehot(Kx16))  -- bit-exact (0/1 B matrix)
typedef __attribute__((ext_vector_type(2))) float v2f;
typedef __attribute__((ext_vector_type(8))) float v8f;

#define QDIM 1000
#define CDIM 91
#define CPAD 92          // pad classes to multiple of 4 (zero column 91)
#define MDIM 300
#define QT   16
#define GEPS 1e-6f

// v_rcp_f32 + one Newton-Raphson step (~0.5 ulp), replaces the ~8-op IEEE div chain
__device__ __forceinline__ float fast_rcp(float d) {
    float r = __builtin_amdgcn_rcpf(d);
    return r * __builtin_fmaf(-d, r, 2.0f);
}

__global__ __launch_bounds__(256)
void matcher_cost_kernel(const float* __restrict__ logits,   // [B,Q,91]
                         const float* __restrict__ pboxes,   // [B,Q,4] cxcywh
                         const int*   __restrict__ tlabels,  // [B,300]
                         const float* __restrict__ tboxes,   // [B,300,4] cxcywh
                         float*       __restrict__ out)      // [B,Q,300]
{
    __shared__ __align__(16) float sProbs[QT * CPAD];   // softmax rows, col 91 = 0
    __shared__ float sPred[QT * 9];                     // cx,cy,w,h,x0,y0,x1,y1,area
    __shared__ float sTgt [MDIM * 9];
    __shared__ int   sLab [MDIM];

    const int b  = blockIdx.y;
    const int q0 = blockIdx.x * QT;
    const int t  = threadIdx.x;

    // ---------- stage 1: softmax of 16 logit rows (16 threads per row) ----------
    {
        const int r = t >> 4;                 // row in tile: threads 0-15 -> r=0 ...
        const int j = t & 15;                 // lane within row (half-wave)
        const int qc = min(q0 + r, QDIM - 1); // clamp: OOB rows finite garbage, never stored
        const float* row = logits + ((size_t)b * QDIM + qc) * CDIM;
        float x[6];
        float mx = -3.0e38f;
#pragma unroll
        for (int i = 0; i < 6; ++i) {
            int c = j + 16 * i;
            x[i] = (c < CDIM) ? row[c] : -3.0e38f;
            mx = fmaxf(mx, x[i]);
        }
#pragma unroll
        for (int off = 8; off >= 1; off >>= 1)           // half-wave reduction
            mx = fmaxf(mx, __shfl_xor(mx, off, 16));
        float s = 0.0f;
#pragma unroll
        for (int i = 0; i < 6; ++i) {
            int c = j + 16 * i;
            x[i] = (c < CDIM) ? __expf(x[i] - mx) : 0.0f;
            s += x[i];
        }
#pragma unroll
        for (int off = 8; off >= 1; off >>= 1)
            s += __shfl_xor(s, off, 16);
        const float inv = fast_rcp(s);
#pragma unroll
        for (int i = 0; i < 6; ++i) {
            int c = j + 16 * i;
            if (c < CDIM) sProbs[r * CPAD + c] = x[i] * inv;
        }
        if (j == 0) sProbs[r * CPAD + CDIM] = 0.0f;      // zero pad column (k=91)
    }

    // ---------- stage 2: pred boxes for this q-tile ----------
    if (t < QT) {
        const int qc = min(q0 + t, QDIM - 1);
        const float* pb = pboxes + ((size_t)b * QDIM + qc) * 4;
        float cx = pb[0], cy = pb[1], w = pb[2], h = pb[3];
        float* d = sPred + t * 9;
        float x0 = cx - 0.5f * w, y0 = cy - 0.5f * h;
        float x1 = cx + 0.5f * w, y1 = cy + 0.5f * h;
        d[0] = cx; d[1] = cy; d[2] = w;  d[3] = h;
        d[4] = x0; d[5] = y0; d[6] = x1; d[7] = y1;
        d[8] = (x1 - x0) * (y1 - y0);
    }

    // ---------- stage 3: all 300 targets ----------
    for (int i = t; i < MDIM; i += 256) {
        sLab[i] = tlabels[(size_t)b * MDIM + i];
        const float* tb = tboxes + ((size_t)b * MDIM + i) * 4;
        float cx = tb[0], cy = tb[1], w = tb[2], h = tb[3];
        float* d = sTgt + i * 9;
        float x0 = cx - 0.5f * w, y0 = cy - 0.5f * h;
        float x1 = cx + 0.5f * w, y1 = cy + 0.5f * h;
        d[0] = cx; d[1] = cy; d[2] = w;  d[3] = h;
        d[4] = x0; d[5] = y0; d[6] = x1; d[7] = y1;
        d[8] = (x1 - x0) * (y1 - y0);
    }
    __syncthreads();

    // ---------- stage 4: per-wave 16x16 output tiles ----------
    const int wave  = t >> 5;
    const int lane  = t & 31;
    const int half  = lane >> 4;     // 0 | 1
    const int ln    = lane & 15;
    const int kpair = half * 2;      // A/B K sub-pair per ISA f32 16x4 layout

    const int nTiles = (MDIM + 15) / 16;          // 19 (last tile 12 cols)
    for (int mt = wave; mt < nTiles; mt += 8) {   // wave-uniform => EXEC all 1s at WMMA
        const int m0     = mt * 16;
        const int mm     = m0 + ln;               // this lane's target column (B/C/D N)
        const bool mOK   = (mm < MDIM);
        const int  mi    = mOK ? mm : (MDIM - 1); // clamped index: no divergent LDS load
        int lab = sLab[mi];
        lab = mOK ? lab : -1;                     // v_cndmask, no EXEC manipulation

        // accumulate probs x onehot over K = 0..91 (23 steps of K=4), exact
        v8f acc = {};
        const float* aRow = sProbs + ln * CPAD + kpair;    // A: M = lane&15
#pragma unroll
        for (int k0 = 0; k0 < CPAD; k0 += 4) {
            v2f a = *(const v2f*)(aRow + k0);              // (K=k0+kpair, +1), 8B aligned
            v2f bm;
            bm.x = (lab == k0 + kpair)     ? 1.0f : 0.0f;  // B: N = lane&15, K mirrors A
            bm.y = (lab == k0 + kpair + 1) ? 1.0f : 0.0f;
            acc = __builtin_amdgcn_wmma_f32_16x16x4_f32(
                false, a, false, bm, (short)0, acc, false, false);
        }

        // this lane's target box
        const float* tg = sTgt + mi * 9;
        const float tcx = tg[0], tcy = tg[1], tw = tg[2], th = tg[3];
        const float tx0 = tg[4], ty0 = tg[5], tx1 = tg[6], ty1 = tg[7], ta = tg[8];

        // acc VGPR v holds (qrow = v + half*8, mcol = mm) per C/D layout
#pragma unroll
        for (int v = 0; v < 8; ++v) {
            const int qrow = v + half * 8;
            const float* pg = sPred + qrow * 9;            // broadcast read across half-wave
            const float l1 = fabsf(pg[0] - tcx) + fabsf(pg[1] - tcy)
                           + fabsf(pg[2] - tw)  + fabsf(pg[3] - th);
            const float px0 = pg[4], py0 = pg[5], px1 = pg[6], py1 = pg[7], pa = pg[8];
            float iw = fminf(px1, tx1) - fmaxf(px0, tx0); iw = fmaxf(iw, 0.0f);
            float ih = fminf(py1, ty1) - fmaxf(py0, ty0); ih = fmaxf(ih, 0.0f);
            const float inter = iw * ih;
            const float uni   = pa + ta - inter;
            const float iou   = inter * fast_rcp(uni + GEPS);
            float cw = fmaxf(px1, tx1) - fminf(px0, tx0); cw = fmaxf(cw, 0.0f);
            float ch = fmaxf(py1, ty1) - fminf(py0, ty0); ch = fmaxf(ch, 0.0f);
            const float ac   = cw * ch;
            const float giou = iou - (ac - uni) * fast_rcp(ac + GEPS);
            const float cost = -acc[v] + 5.0f * l1 - 2.0f * giou;  // 1*cls + 5*L1 + 2*(-giou)

            const int q = q0 + qrow;
            if (q < QDIM && mOK)
                out[((size_t)b * QDIM + q) * MDIM + mm] = cost;
        }
    }
}

extern "C" void kernel_launch(void* const* d_in, const int* in_sizes, int n_in,
                              void* d_out, int out_size, void* d_ws, size_t ws_size,
                              hipStream_t stream) {
    const float* logits  = (const float*)d_in[0];   // [B,1000,91]
    const float* pboxes  = (const float*)d_in[1];   // [B,1000,4]
    const int*   tlabels = (const int*)  d_in[2];   // [B,300]
    const float* tboxes  = (const float*)d_in[3];   // [B,300,4]
    float*       out     = (float*)d_out;           // [B,1000,300]

    const int B = in_sizes[2] / MDIM;               // 64
    dim3 grid((QDIM + QT - 1) / QT, B);             // 63 x 64 blocks
    dim3 block(256);                                // 8 waves (wave32)
    matcher_cost_kernel<<<grid, block, 0, stream>>>(logits, pboxes, tlabels, tboxes, out);
}